// GraphAttentionLayerV2_42039139893266
// MI455X (gfx1250) — compile-verified
//
#include <hip/hip_runtime.h>

// GATv2 (eval) for MI455X / gfx1250.
// B=8, N=512, F=128, H=4, U=32, leaky_alpha=0.2
//
// Stage 1 (gat2_proj_kernel): hsrc/hdst = inputs @ W_{src,dst}[h]
//   via V_WMMA_F32_16X16X4_F32, one wave per 16x16 output tile.
// Stage 2 (gat2_attn_kernel): per (b,h,i-tile of 16):
//   scores (VALU, operands LDS-staged) -> softmax (LDS) ->
//   attn @ h_src via V_WMMA_F32_16X16X4_F32 (A from LDS, B from global).

#define B_ 8
#define N_ 512
#define F_ 128
#define H_ 4
#define U_ 32
#define ALPHA 0.2f

typedef float v2f __attribute__((ext_vector_type(2)));
typedef float v8f __attribute__((ext_vector_type(8)));

// ---------------------------------------------------------------------------
// Kernel 1: projections.  grid = (N/16, U/16, 2*H*B), block = 32 (one wave).
// hsrc[h][b][i][u] = sum_k inputs[b][i][k] * W_src[h][k][u]   (same for dst)
// ---------------------------------------------------------------------------
__global__ __launch_bounds__(32)
void gat2_proj_kernel(const float* __restrict__ inputs,
                      const float* __restrict__ W_src,
                      const float* __restrict__ W_dst,
                      float* __restrict__ hsrc,
                      float* __restrict__ hdst)
{
    const int lane = threadIdx.x;      // 0..31
    const int m    = lane & 15;        // row (A) / col (B,C)
    const int half = lane >> 4;        // K-half selector per ISA layout

    const int tile_i = blockIdx.x;     // N/16 tiles of rows
    const int tile_u = blockIdx.y;     // U/16 tiles of cols
    const int z      = blockIdx.z;     // (proj*H + h)*B + b
    const int b      = z % B_;
    const int hh     = z / B_;
    const int h      = hh % H_;
    const int proj   = hh / H_;        // 0 = src, 1 = dst

    const float* W   = (proj ? W_dst : W_src) + (size_t)h * F_ * U_;
    float*       out = (proj ? hdst : hsrc) + ((size_t)(h * B_ + b) * N_) * U_;

    const float* arow = inputs + ((size_t)(b * N_) + tile_i * 16 + m) * F_;

    // A 16x4 f32 layout: lane m holds row m; VGPR v holds K = k0 + v,
    // k0 = 4*step + 2*half.  B 4x16: mirrored K-split across lane halves.
    v8f acc = {};
    #pragma unroll 4
    for (int step = 0; step < F_ / 4; ++step) {
        const int k0 = 4 * step + 2 * half;
        v2f av, bv;
        av.x = arow[k0];
        av.y = arow[k0 + 1];
        bv.x = W[(size_t)(k0    ) * U_ + tile_u * 16 + m];
        bv.y = W[(size_t)(k0 + 1) * U_ + tile_u * 16 + m];
        acc = __builtin_amdgcn_wmma_f32_16x16x4_f32(false, av, false, bv,
                                                    (short)0, acc, false, false);
    }

    // C/D 16x16 f32: VGPR v holds M = v + 8*half, N = lane&15
    #pragma unroll
    for (int v = 0; v < 8; ++v) {
        const int row = tile_i * 16 + v + 8 * half;
        out[(size_t)row * U_ + tile_u * 16 + m] = acc[v];
    }
}

// ---------------------------------------------------------------------------
// Kernel 2: scores + softmax + attn@h_src.
// grid = (N/16, B*H), block = 128 (4 waves).
// ---------------------------------------------------------------------------
__global__ __launch_bounds__(128)
void gat2_attn_kernel(const float* __restrict__ hsrc,
                      const float* __restrict__ hdst,
                      const float* __restrict__ avec,   // (H, U)
                      float* __restrict__ out)          // (B, N, H*U)
{
    __shared__ float s_scores[16 * N_];   // 32 KB attn block (16 rows x 512)
    __shared__ float s_srct[16 * U_];     // 2 KB: the 16 query rows of h_src
    __shared__ float s_a[U_];
    __shared__ float s_inv[16];

    const int t      = threadIdx.x;      // 0..127
    const int tile_i = blockIdx.x;       // query-row tile
    const int bh     = blockIdx.y;
    const int b      = bh / H_;
    const int h      = bh % H_;

    const float* srcb = hsrc + ((size_t)(h * B_ + b) * N_) * U_;
    const float* dstb = hdst + ((size_t)(h * B_ + b) * N_) * U_;

    // stage query tile + attention vector into LDS
    for (int idx = t; idx < 16 * U_; idx += 128)
        s_srct[idx] = srcb[(size_t)(tile_i * 16) * U_ + idx];
    if (t < U_) s_a[t] = avec[h * U_ + t];
    __syncthreads();

    // scores[m][j] = sum_u a[u] * leaky_relu(src[m][u] + dst[j][u])
    for (int idx = t; idx < 16 * N_; idx += 128) {
        const int m = idx >> 9;          // idx / 512
        const int j = idx & (N_ - 1);
        const float* dj = dstb + (size_t)j * U_;
        float s = 0.f;
        #pragma unroll
        for (int u = 0; u < U_; ++u) {
            const float zv = s_srct[m * U_ + u] + dj[u];
            s = fmaf(s_a[u], (zv > 0.f ? zv : ALPHA * zv), s);
        }
        s_scores[idx] = s;
    }
    __syncthreads();

    // softmax over j: one row per thread (rows 0..15)
    if (t < 16) {
        float mx = -3.402823466e38f;
        for (int j = 0; j < N_; ++j) mx = fmaxf(mx, s_scores[t * N_ + j]);
        float sum = 0.f;
        for (int j = 0; j < N_; ++j) {
            const float e = __expf(s_scores[t * N_ + j] - mx);
            s_scores[t * N_ + j] = e;
            sum += e;
        }
        s_inv[t] = 1.f / sum;
    }
    __syncthreads();
    for (int idx = t; idx < 16 * N_; idx += 128)
        s_scores[idx] *= s_inv[idx >> 9];
    __syncthreads();

    // out_tile(16 x 32) = attn(16 x 512) @ h_src(512 x 32)
    // waves 0,1 each own one 16-wide slice of U; K=512 -> 128 WMMA steps.
    const int wave = t >> 5;
    const int lane = t & 31;
    if (wave < 2) {
        const int ntile = wave;
        const int m     = lane & 15;
        const int half  = lane >> 4;
        v8f acc = {};
        for (int step = 0; step < N_ / 4; ++step) {
            const int k0 = 4 * step + 2 * half;
            v2f av, bv;
            av.x = s_scores[m * N_ + k0];        // A from LDS (ds_load)
            av.y = s_scores[m * N_ + k0 + 1];
            bv.x = srcb[(size_t)(k0    ) * U_ + ntile * 16 + m];
            bv.y = srcb[(size_t)(k0 + 1) * U_ + ntile * 16 + m];
            acc = __builtin_amdgcn_wmma_f32_16x16x4_f32(false, av, false, bv,
                                                        (short)0, acc, false, false);
        }
        #pragma unroll
        for (int v = 0; v < 8; ++v) {
            const int row = tile_i * 16 + v + 8 * half;
            out[((size_t)(b * N_) + row) * (H_ * U_) + h * U_ + ntile * 16 + m] = acc[v];
        }
    }
}

// ---------------------------------------------------------------------------
extern "C" void kernel_launch(void* const* d_in, const int* in_sizes, int n_in,
                              void* d_out, int out_size, void* d_ws, size_t ws_size,
                              hipStream_t stream) {
    (void)in_sizes; (void)n_in; (void)out_size; (void)ws_size;

    const float* inputs = (const float*)d_in[0];  // (B, N, F)
    const float* W_src  = (const float*)d_in[1];  // (H, F, U)
    const float* W_dst  = (const float*)d_in[2];  // (H, F, U)
    const float* avec   = (const float*)d_in[3];  // (H, U, 1)
    float*       out    = (float*)d_out;          // (B, N, H*U)

    float* hsrc = (float*)d_ws;                               // H*B*N*U floats
    float* hdst = hsrc + (size_t)H_ * B_ * N_ * U_;           // H*B*N*U floats

    dim3 g1(N_ / 16, U_ / 16, 2 * H_ * B_);
    gat2_proj_kernel<<<g1, dim3(32), 0, stream>>>(inputs, W_src, W_dst, hsrc, hdst);

    dim3 g2(N_ / 16, B_ * H_);
    gat2_attn_kernel<<<g2, dim3(128), 0, stream>>>(hsrc, hdst, avec, out);
}